// IGCN_43808666419475
// MI455X (gfx1250) — compile-verified
//
#include <hip/hip_runtime.h>

typedef float v2f __attribute__((ext_vector_type(2)));
typedef float v8f __attribute__((ext_vector_type(8)));

#define D_FEAT    256
#define D_HID     256
#define N_CLASSES 64
#define ROWS_PER_BLOCK 128
#define HS_PITCH  (D_HID + 2)   // pad 2 dwords: DS bank spread for strided row reads

// ---------------------------------------------------------------------------
// Fused 2-layer MLP: x = relu(X @ W1 + b1) @ W2 + b2, written to xout [N,64].
// Block = 256 threads (8 waves, wave32). Wave w owns node rows [w*16, w*16+16).
// Register-blocked fp32 WMMA (V_WMMA_F32_16X16X4_F32):
//   GEMM1: 16 v8f accumulators per wave (full 16x256 h tile in VGPRs);
//          per K-step one b64 A load feeds 16 WMMAs.
//   GEMM2: 4 v8f accumulators; per K-step one ds_load_b64 feeds 4 WMMAs.
//  A 16x4 layout:  lane<16 -> M=lane,    K = kt*4 + {0,1}
//                  lane>=16 -> M=lane-16, K = kt*4 + {2,3}
//  B 4x16 layout:  lane%16 = N, same K split per half-wave
//  C/D 16x16:      lane%16 = N, vgpr r -> M = r + (lane>=16 ? 8 : 0)
// ---------------------------------------------------------------------------
__global__ __launch_bounds__(256) void IGCN_mlp_wmma(
    const float* __restrict__ X,
    const float* __restrict__ W1, const float* __restrict__ b1,
    const float* __restrict__ W2, const float* __restrict__ b2,
    float* __restrict__ xout, int nNodes)
{
    extern __shared__ float Hs[];   // [ROWS_PER_BLOCK][HS_PITCH]

    const int tid   = threadIdx.x;
    const int wave  = tid >> 5;
    const int lane  = tid & 31;
    const int lhalf = lane & 15;
    const int khalf = (lane >> 4) ? 2 : 0;   // K offset for upper half-wave
    const int rhalf = (lane >> 4) ? 8 : 0;   // C/D row offset for upper half-wave
    const int base  = blockIdx.x * ROWS_PER_BLOCK;

    // ---------------- GEMM1: h = relu(X @ W1 + b1) -> LDS ----------------
    {
        int mrow = base + wave * 16 + lhalf;
        if (mrow >= nNodes) mrow = nNodes - 1;            // tail clamp (stores guarded later)
        const float* __restrict__ xrow = X + (size_t)mrow * D_FEAT;

        v8f acc[D_HID / 16];                              // 16 x v8f = 128 VGPRs
        #pragma unroll
        for (int nt = 0; nt < D_HID / 16; ++nt) acc[nt] = (v8f){};

        for (int kt = 0; kt < D_FEAT / 4; ++kt) {
            const int k0 = kt * 4 + khalf;
            v2f a = *(const v2f*)(xrow + k0);             // one b64 load, reused 16x
            const float* __restrict__ w1p = W1 + (size_t)k0 * D_HID + lhalf;
            #pragma unroll
            for (int nt = 0; nt < D_HID / 16; ++nt) {
                v2f b;
                b.x = w1p[nt * 16];                       // L2/L1-hot (W1 = 256 KB)
                b.y = w1p[D_HID + nt * 16];
                acc[nt] = __builtin_amdgcn_wmma_f32_16x16x4_f32(
                            false, a, false, b, (short)0, acc[nt], false, false);
            }
        }

        const int rbase = wave * 16 + rhalf;
        #pragma unroll
        for (int nt = 0; nt < D_HID / 16; ++nt) {
            const int n = nt * 16 + lhalf;
            const float bias = b1[n];
            #pragma unroll
            for (int r = 0; r < 8; ++r) {
                float h = acc[nt][r] + bias;
                h = h > 0.f ? h : 0.f;                    // ReLU
                Hs[(rbase + r) * HS_PITCH + n] = h;       // lanes 0-15 contiguous: no DS conflicts
            }
        }
    }

    __syncthreads();

    // ---------------- GEMM2: x = h @ W2 + b2 -> global ----------------
    {
        const float* __restrict__ hrow = &Hs[(wave * 16 + lhalf) * HS_PITCH];

        v8f acc[N_CLASSES / 16];                          // 4 x v8f
        #pragma unroll
        for (int nt = 0; nt < N_CLASSES / 16; ++nt) acc[nt] = (v8f){};

        for (int kt = 0; kt < D_HID / 4; ++kt) {
            const int k0 = kt * 4 + khalf;
            v2f a = *(const v2f*)(hrow + k0);             // one ds_load_b64, reused 4x
            const float* __restrict__ w2p = W2 + (size_t)k0 * N_CLASSES + lhalf;
            #pragma unroll
            for (int nt = 0; nt < N_CLASSES / 16; ++nt) {
                v2f b;
                b.x = w2p[nt * 16];                       // L2-hot (W2 = 64 KB)
                b.y = w2p[N_CLASSES + nt * 16];
                acc[nt] = __builtin_amdgcn_wmma_f32_16x16x4_f32(
                            false, a, false, b, (short)0, acc[nt], false, false);
            }
        }

        const int rbase = wave * 16 + rhalf;
        #pragma unroll
        for (int nt = 0; nt < N_CLASSES / 16; ++nt) {
            const int n = nt * 16 + lhalf;
            const float bias = b2[n];
            #pragma unroll
            for (int r = 0; r < 8; ++r) {
                const int node = base + rbase + r;
                if (node < nNodes)
                    xout[(size_t)node * N_CLASSES + n] = acc[nt][r] + bias;
            }
        }
    }
}

// ---------------------------------------------------------------------------
// Zero logits buffer (harness poisons d_out; atomics accumulate into it).
// ---------------------------------------------------------------------------
__global__ __launch_bounds__(256) void IGCN_zero(float4* __restrict__ out, int n4)
{
    int i = blockIdx.x * blockDim.x + threadIdx.x;
    if (i < n4) {
        float4 z = {0.f, 0.f, 0.f, 0.f};
        out[i] = z;
    }
}

// ---------------------------------------------------------------------------
// COO scatter: out[row[e]] += vals[e] * x[col[e]].
// One 32-lane wave handles one edge; lane covers classes {lane, lane+32}.
// x (25.6 MB) and out (25.6 MB) are both L2-resident on MI455X (192 MB L2),
// so gathers and f32 atomics stay on-chip.
// ---------------------------------------------------------------------------
__global__ __launch_bounds__(256) void IGCN_scatter(
    const int* __restrict__ row, const int* __restrict__ col,
    const float* __restrict__ vals, const float* __restrict__ x,
    float* __restrict__ out, int nEdges)
{
    const int t = blockIdx.x * blockDim.x + threadIdx.x;
    const int e = t >> 5;
    const int l = t & 31;
    if (e >= nEdges) return;

    const int   r = row[e];
    const int   c = col[e];
    const float v = vals[e];

    const float g0 = v * x[(size_t)c * N_CLASSES + l];        // 128B coalesced gather
    const float g1 = v * x[(size_t)c * N_CLASSES + 32 + l];
    atomicAdd(out + (size_t)r * N_CLASSES + l,      g0);      // global_atomic_add_f32
    atomicAdd(out + (size_t)r * N_CLASSES + 32 + l, g1);
}

extern "C" void kernel_launch(void* const* d_in, const int* in_sizes, int n_in,
                              void* d_out, int out_size, void* d_ws, size_t ws_size,
                              hipStream_t stream)
{
    const float* X    = (const float*)d_in[0];
    const int*   row  = (const int*)  d_in[1];
    const int*   col  = (const int*)  d_in[2];
    const float* ev   = (const float*)d_in[3];
    const float* W1   = (const float*)d_in[4];
    const float* b1   = (const float*)d_in[5];
    const float* W2   = (const float*)d_in[6];
    const float* b2   = (const float*)d_in[7];
    float*       out  = (float*)d_out;

    const int nNodes = in_sizes[0] / D_FEAT;
    const int nEdges = in_sizes[1];
    float* xbuf = (float*)d_ws;                 // [nNodes, 64] logits scratch

    // 1) fused MLP -> xbuf
    const int blocks1 = (nNodes + ROWS_PER_BLOCK - 1) / ROWS_PER_BLOCK;
    const size_t lds  = sizeof(float) * ROWS_PER_BLOCK * HS_PITCH;
    IGCN_mlp_wmma<<<blocks1, 256, lds, stream>>>(X, W1, b1, W2, b2, xbuf, nNodes);

    // 2) zero output
    const int n4 = out_size / 4;
    IGCN_zero<<<(n4 + 255) / 256, 256, 0, stream>>>((float4*)out, n4);

    // 3) edge gather + atomic scatter-add
    const long long work = (long long)nEdges * 32;
    const int blocks3 = (int)((work + 255) / 256);
    IGCN_scatter<<<blocks3, 256, 0, stream>>>(row, col, ev, xbuf, out, nEdges);
}